// Linear_57337813402049
// MI455X (gfx1250) — compile-verified
//
#include <hip/hip_runtime.h>

typedef __attribute__((ext_vector_type(2))) float v2f;
typedef __attribute__((ext_vector_type(8))) float v8f;

// Equivariant block-diagonal linear, one irrep block per launch.
// Y[m, w] = sum_u A[m, u] * W[u, w], where m = n*D + i and
//   A[m, u] = x[n*1056 + OFF + u*D + i]
//   Y[m, w] -> out[n*1056 + OFF + w*D + i]
//
// One wave per 16-row m-tile: A panel lives in registers (x read once from
// HBM). W is transposed into LDS once per workgroup with an XOR bank swizzle,
// so each WMMA's weight pair (u, u+1) for a lane is a single conflict-free
// ds_load_b64 — weight traffic moves off the VMEM pipe entirely.
// WMMA f32 16x16x4 layout: lane&15 = A row / B,D column; lane>>4 selects the
// K pair {0,1} vs {2,3} inside each K-group of 4.
template <int MUL, int D, int OFF>
__global__ __launch_bounds__(256) void irrep_linear_kernel(
    const float* __restrict__ x, const float* __restrict__ wgt,
    float* __restrict__ out, int ntiles)
{
    constexpr int KSTEPS = MUL / 4;   // K-groups of 4 (32, 32, 16, 8 -> even)
    constexpr int WTILES = MUL / 16;  // output column tiles
    constexpr int KCHUNK = (KSTEPS < 16) ? KSTEPS : 16;  // B staged per chunk

    // XOR swizzle (multiple of 4) keeps the 32 lanes' 8B reads on 64 distinct
    // banks while preserving (u, u+1) adjacency and 8B alignment.
    auto swz = [](int wc) -> int {
        return (MUL >= 64) ? (4 * (wc & 15)) : (4 * ((wc >> 1) & 7));
    };

    // --- Transpose W into LDS (whole workgroup, before any wave exits) ----
    __shared__ float wT[MUL * MUL];
    for (int e = threadIdx.x; e < MUL * MUL; e += 256) {
        const int u  = e / MUL;       // row of W (contraction index)
        const int wc = e - u * MUL;   // column of W (output index)
        wT[wc * MUL + (u ^ swz(wc))] = wgt[e];
    }
    __syncthreads();

    const int lane = threadIdx.x & 31;
    const int wave = threadIdx.x >> 5;
    const int tile = blockIdx.x * 8 + wave;   // 8 waves per 256-thread block
    if (tile >= ntiles) return;               // wave-uniform: EXEC stays all-1s

    const int half = lane >> 4;   // 0: K={0,1}, 1: K={2,3} within a K-group
    const int lidx = lane & 15;   // A row / B,D column within the tile

    // --- Load the A panel for this lane's row, once, into registers -------
    const int m = tile * 16 + lidx;     // flattened (n, i) row, i fastest
    const int n = m / D;
    const int i = m - n * D;
    const int rowbase = n * 1056 + OFF + i;  // + u*D addresses element u

    float a0[KSTEPS], a1[KSTEPS];
#pragma unroll
    for (int ks = 0; ks < KSTEPS; ++ks) {
        const int u0 = 4 * ks + 2 * half;
        a0[ks] = x[rowbase + u0 * D];
        a1[ks] = x[rowbase + (u0 + 1) * D];
    }

    const int mbase = tile * 16;

    // --- Sweep output-column tiles, reusing the register-resident A -------
#pragma unroll
    for (int wt = 0; wt < WTILES; ++wt) {
        const int wcol = wt * 16 + lidx;   // this lane's output column
        const int brow = wcol * MUL;       // LDS row base for this column
        const int bswz = swz(wcol);

        v8f c0 = {0.f, 0.f, 0.f, 0.f, 0.f, 0.f, 0.f, 0.f};
        v8f c1 = {0.f, 0.f, 0.f, 0.f, 0.f, 0.f, 0.f, 0.f};

#pragma unroll
        for (int kc = 0; kc < KSTEPS; kc += KCHUNK) {
            // Stage weight pairs: one conflict-free ds_load_b64 per K-group,
            // distinct destination regs -> single amortized DScnt wait.
            v2f bb[KCHUNK];
#pragma unroll
            for (int j = 0; j < KCHUNK; ++j) {
                const int u0 = 4 * (kc + j) + 2 * half;
                bb[j] = *(const v2f*)&wT[brow + (u0 ^ bswz)];
            }
            // Back-to-back WMMA burst on the staged operands.
#pragma unroll
            for (int j = 0; j < KCHUNK; j += 2) {
                {
                    v2f a = {a0[kc + j], a1[kc + j]};
                    c0 = __builtin_amdgcn_wmma_f32_16x16x4_f32(
                            false, a, false, bb[j], (short)0, c0, false, false);
                }
                {
                    v2f a = {a0[kc + j + 1], a1[kc + j + 1]};
                    c1 = __builtin_amdgcn_wmma_f32_16x16x4_f32(
                            false, a, false, bb[j + 1], (short)0, c1, false, false);
                }
            }
        }
        const v8f c = c0 + c1;

        // --- Store: VGPR r holds row (r + 8*half), column wcol ------------
#pragma unroll
        for (int r = 0; r < 8; ++r) {
            const int ms = mbase + r + 8 * half;
            const int ns = ms / D;
            const int is = ms - ns * D;
            out[ns * 1056 + OFF + wcol * D + is] = c[r];
        }
    }
}

extern "C" void kernel_launch(void* const* d_in, const int* in_sizes, int n_in,
                              void* d_out, int out_size, void* d_ws, size_t ws_size,
                              hipStream_t stream) {
    const float* x  = (const float*)d_in[0];
    const float* w0 = (const float*)d_in[1];
    const float* w1 = (const float*)d_in[2];
    const float* w2 = (const float*)d_in[3];
    const float* w3 = (const float*)d_in[4];
    float* out = (float*)d_out;

    const int nnodes = in_sizes[0] / 1056;  // DIM = 1056

    // 128x0e : MUL=128, D=1, OFF=0
    {
        const int ntiles = (nnodes * 1) / 16;
        const int blocks = (ntiles + 7) / 8;
        irrep_linear_kernel<128, 1, 0><<<blocks, 256, 0, stream>>>(x, w0, out, ntiles);
    }
    // 128x1o : MUL=128, D=3, OFF=128
    {
        const int ntiles = (nnodes * 3) / 16;
        const int blocks = (ntiles + 7) / 8;
        irrep_linear_kernel<128, 3, 128><<<blocks, 256, 0, stream>>>(x, w1, out, ntiles);
    }
    // 64x2e : MUL=64, D=5, OFF=512
    {
        const int ntiles = (nnodes * 5) / 16;
        const int blocks = (ntiles + 7) / 8;
        irrep_linear_kernel<64, 5, 512><<<blocks, 256, 0, stream>>>(x, w2, out, ntiles);
    }
    // 32x3o : MUL=32, D=7, OFF=832
    {
        const int ntiles = (nnodes * 7) / 16;
        const int blocks = (ntiles + 7) / 8;
        irrep_linear_kernel<32, 7, 832><<<blocks, 256, 0, stream>>>(x, w3, out, ntiles);
    }
}